// FasterRCNN_56521769615800
// MI455X (gfx1250) — compile-verified
//
#include <hip/hip_runtime.h>
#include <hip/hip_bf16.h>
#include <math.h>

#define NCLS      20
#define RBOX      2000
#define NSORT     2048
#define CHUNK     256
#define SCORE_THR 0.05f
#define NMS_THR   0.3f

#if defined(__has_builtin)
#if __has_builtin(__builtin_amdgcn_tensor_load_to_lds)
#define HAVE_TDM 1
#endif
#endif
#ifndef HAVE_TDM
#define HAVE_TDM 0
#endif

typedef unsigned int u32x4 __attribute__((ext_vector_type(4)));
typedef int          i32x4 __attribute__((ext_vector_type(4)));
typedef int          i32x8 __attribute__((ext_vector_type(8)));

__device__ __forceinline__ unsigned lds_off(const void* p) {
  // Low 32 bits of the flat LDS-aperture address == wave-relative LDS byte offset (ISA 10.2)
  return (unsigned)(size_t)p;
}

__global__ __launch_bounds__(1024, 1)
void frcnn_detect_kernel(const float* __restrict__ boxes,
                         const float* __restrict__ loc,
                         const float* __restrict__ score,
                         const int* __restrict__ ihp,
                         const int* __restrict__ iwp,
                         float* __restrict__ out) {
  __shared__ float    sraw[CHUNK * 4];              // async-staged raw boxes (chunked)
  __shared__ float    sloc[CHUNK * 4];              // TDM-staged per-class loc deltas (chunked)
  __shared__ float    uby0[NSORT], ubx0[NSORT];     // decoded clipped boxes, by ROI index
  __shared__ float    uby1[NSORT], ubx1[NSORT];
  __shared__ float    skey[NSORT];                  // sort key (prob if valid else -inf)
  __shared__ int      sidx[NSORT];                  // ROI index payload
  __shared__ unsigned char skeep[NSORT];            // keep flag per sorted position
  __shared__ unsigned sup[NSORT / 32];              // suppression bitmask
  __shared__ int      s_nv;                         // number of valid (prob > thr) entries

  const int   t = threadIdx.x;
  const int   c = blockIdx.x;                       // foreground class c -> raw class c+1
  const float NEG_INF = __int_as_float((int)0xff800000u);
  const float imgH = (float)*ihp;
  const float imgW = (float)*iwp;

  // ---- init LDS ----
  for (int p = t; p < NSORT; p += blockDim.x) {
    skey[p] = NEG_INF;
    sidx[p] = p;
    uby0[p] = 0.f; ubx0[p] = 0.f; uby1[p] = 0.f; ubx1[p] = 0.f;
    skeep[p] = 0;
  }
  if (t < NSORT / 32) sup[t] = 0u;
  if (t == 0) s_nv = 0;
  __syncthreads();

  // ---- Phase A: chunked staging (per-lane ASYNC for boxes, TDM for strided loc) + decode ----
  for (int base = 0; base < RBOX; base += CHUNK) {
    const int n = min(CHUNK, RBOX - base);

    // contiguous raw boxes: per-lane async global->LDS b128 (ASYNCcnt path)
    if (t < n) {
      unsigned           lo = lds_off(&sraw[t * 4]);
      unsigned long long ga = (unsigned long long)(const void*)(boxes + (size_t)(base + t) * 4);
      asm volatile("global_load_async_to_lds_b128 %0, %1, off"
                   :: "v"(lo), "v"(ga) : "memory");
    }

#if HAVE_TDM
    // strided loc gather (16B of every 336B row): one TDM descriptor per chunk (TENSORcnt path)
    if (t < 32) {
      const unsigned long long ga =
          (unsigned long long)(const void*)(loc + (size_t)base * (4 * (NCLS + 1)) +
                                            (size_t)(c + 1) * 4);
      u32x4 g0;
      g0[0] = 0x1u;                                             // count=1, user descriptor
      g0[1] = lds_off(sloc);                                    // LDS dest byte address
      g0[2] = (unsigned)ga;                                     // global_addr[31:0]
      g0[3] = ((unsigned)(ga >> 32) & 0x01FFFFFFu)              // global_addr[56:32]
              | 0x80000000u;                                    // type=2 ("image")
      i32x8 g1;
      g1[0] = 0x00020000;                                       // wg_mask=0, data_size=4B
      g1[1] = (int)(84u << 16);                                 // tensor_dim0 = 84 elems/row
      g1[2] = (int)((unsigned)n << 16);                         // tensor_dim1 = n rows
      g1[3] = (int)(4u << 16);                                  // tile_dim0 = 4 elems
      g1[4] = n;                                                // tile_dim1 = n, tile_dim2 = 0
      g1[5] = 84;                                               // tensor_dim0_stride[31:0]
      g1[6] = 0;
      g1[7] = 0;
      i32x4 z4 = {0, 0, 0, 0};
#if __clang_major__ >= 23
      i32x8 z8 = {0, 0, 0, 0, 0, 0, 0, 0};
      __builtin_amdgcn_tensor_load_to_lds(g0, g1, z4, z4, z8, 0);
#else
      __builtin_amdgcn_tensor_load_to_lds(g0, g1, z4, z4, 0);
#endif
      __builtin_amdgcn_s_wait_tensorcnt(0);
    }
#else
    // fallback: per-lane async gather of the class's loc float4
    if (t < n) {
      unsigned           lo = lds_off(&sloc[t * 4]);
      unsigned long long ga = (unsigned long long)(const void*)(
          loc + (size_t)(base + t) * (4 * (NCLS + 1)) + (size_t)(c + 1) * 4);
      asm volatile("global_load_async_to_lds_b128 %0, %1, off"
                   :: "v"(lo), "v"(ga) : "memory");
    }
#endif
    asm volatile("s_wait_asynccnt 0x0" ::: "memory");
    __syncthreads();

    if (t < n) {
      const int r  = base + t;
      const float y1 = sraw[t * 4 + 0], x1 = sraw[t * 4 + 1];
      const float y2 = sraw[t * 4 + 2], x2 = sraw[t * 4 + 3];
      const float sh  = y2 - y1,          sw  = x2 - x1;
      const float scy = y1 + 0.5f * sh,   scx = x1 + 0.5f * sw;

      const float dy = sloc[t * 4 + 0], dx = sloc[t * 4 + 1];
      const float dh = sloc[t * 4 + 2], dw = sloc[t * 4 + 3];
      const float cy = dy * sh + scy;
      const float cx = dx * sw + scx;
      const float hh = expf(dh) * sh;
      const float ww = expf(dw) * sw;

      const float oy1 = fminf(fmaxf(cy - 0.5f * hh, 0.f), imgH);
      const float ox1 = fminf(fmaxf(cx - 0.5f * ww, 0.f), imgW);
      const float oy2 = fminf(fmaxf(cy + 0.5f * hh, 0.f), imgH);
      const float ox2 = fminf(fmaxf(cx + 0.5f * ww, 0.f), imgW);

      // softmax over 21 scores, probability of class c+1
      const float* srow = score + (size_t)r * (NCLS + 1);
      float m = srow[0];
      #pragma unroll
      for (int k = 1; k < NCLS + 1; ++k) m = fmaxf(m, srow[k]);
      float ssum = 0.f;
      #pragma unroll
      for (int k = 0; k < NCLS + 1; ++k) ssum += expf(srow[k] - m);
      const float prob = expf(srow[c + 1] - m) / ssum;

      uby0[r] = oy1; ubx0[r] = ox1; uby1[r] = oy2; ubx1[r] = ox2;
      const bool valid = (prob > SCORE_THR);
      skey[r] = valid ? prob : NEG_INF;
      if (valid) atomicAdd(&s_nv, 1);
    }
    __syncthreads();
  }

  // ---- Phase B: bitonic sort (key descending, index payload), 2048 elems / 1024 threads ----
  for (int k = 2; k <= NSORT; k <<= 1) {
    for (int j = k >> 1; j > 0; j >>= 1) {
      const int  i  = ((t & ~(j - 1)) << 1) | (t & (j - 1));
      const int  p  = i | j;
      const bool up = ((i & k) == 0);
      const float ki = skey[i], kp = skey[p];
      const bool  sw = up ? (ki < kp) : (ki > kp);
      if (sw) {
        skey[i] = kp; skey[p] = ki;
        const int ti = sidx[i]; sidx[i] = sidx[p]; sidx[p] = ti;
      }
      __syncthreads();
    }
  }

  // ---- Phase C: exact sequential greedy NMS over valid prefix, single wave32 ----
  // All valid entries (key > thr) sort before all -inf entries; invalid ones can neither
  // be kept nor suppress anything, so the scan/sweep only needs positions [0, nv).
  const int nv = s_nv;
  if (t < 32) {
    const int lane = t;
    for (int i = 0; i < nv; ++i) {
      const unsigned w      = sup[i >> 5];
      const bool     keep_i = (((w >> (i & 31)) & 1u) == 0u);
      if (lane == 0) skeep[i] = keep_i ? 1 : 0;
      if (keep_i) {
        const int   ri    = sidx[i];
        const float iy1 = uby0[ri], ix1 = ubx0[ri], iy2 = uby1[ri], ix2 = ubx1[ri];
        const float iarea = (iy2 - iy1) * (ix2 - ix1);
        for (int j = i + 1 + lane; j < nv; j += 32) {
          const int   rj = sidx[j];
          const float jy1 = uby0[rj], jx1 = ubx0[rj], jy2 = uby1[rj], jx2 = ubx1[rj];
          const float ty = fmaxf(iy1, jy1), tx = fmaxf(ix1, jx1);
          const float by = fminf(iy2, jy2), bx = fminf(ix2, jx2);
          const float inter = fmaxf(by - ty, 0.f) * fmaxf(bx - tx, 0.f);
          const float ja    = (jy2 - jy1) * (jx2 - jx1);
          const float uni   = iarea + ja - inter;
          const float iou   = inter / fmaxf(uni, 1e-9f);
          if (iou > NMS_THR) atomicOr(&sup[j >> 5], 1u << (j & 31));
        }
      }
      __builtin_amdgcn_wave_barrier();   // keep compiler from moving LDS ops across iterations
    }
  }
  __syncthreads();

  // ---- Phase D: write (boxes|prob)*keep, labels*keep, keep ----
  float* outBoxes  = out;                                       // (20,2000,5)
  float* outLabels = out + (size_t)NCLS * RBOX * 5;             // (20,2000)
  float* outKeep   = outLabels + (size_t)NCLS * RBOX;           // (20,2000)
  for (int p = t; p < RBOX; p += blockDim.x) {
    const bool  kp = (skeep[p] != 0);
    const int   ri = sidx[p];
    const size_t ob = ((size_t)c * RBOX + p) * 5;
    outBoxes[ob + 0] = kp ? uby0[ri] : 0.f;
    outBoxes[ob + 1] = kp ? ubx0[ri] : 0.f;
    outBoxes[ob + 2] = kp ? uby1[ri] : 0.f;
    outBoxes[ob + 3] = kp ? ubx1[ri] : 0.f;
    outBoxes[ob + 4] = kp ? skey[p]  : 0.f;
    outLabels[(size_t)c * RBOX + p] = kp ? (float)(c + 1) : 0.f;
    outKeep  [(size_t)c * RBOX + p] = kp ? 1.f : 0.f;
  }
}

extern "C" void kernel_launch(void* const* d_in, const int* in_sizes, int n_in,
                              void* d_out, int out_size, void* d_ws, size_t ws_size,
                              hipStream_t stream) {
  (void)in_sizes; (void)n_in; (void)d_ws; (void)ws_size; (void)out_size;
  const float* boxes = (const float*)d_in[0];  // (2000,4)
  const float* loc   = (const float*)d_in[1];  // (2000,84)
  const float* score = (const float*)d_in[2];  // (2000,21)
  const int*   ihp   = (const int*)d_in[3];    // scalar 600
  const int*   iwp   = (const int*)d_in[4];    // scalar 800
  float*       outp  = (float*)d_out;

  frcnn_detect_kernel<<<dim3(NCLS), dim3(1024), 0, stream>>>(boxes, loc, score, ihp, iwp, outp);
}